// GATGRU_54322746359883
// MI455X (gfx1250) — compile-verified
//
#include <hip/hip_runtime.h>
#include <hip/hip_bf16.h>
#include <stdint.h>

#define HEADS     3
#define C_DIM     384
#define GAT_IN    24
#define HID       32
#define T_HIS     12
#define HC        (HEADS * C_DIM)   // 1152
#define NEG_SLOPE 0.2f

typedef float v2f __attribute__((ext_vector_type(2)));
typedef float v8f __attribute__((ext_vector_type(8)));

// ---------- helpers: order-preserving float<->uint for atomic max ----------
__device__ __forceinline__ unsigned encf(float f) {
    unsigned u = __float_as_uint(f);
    return u ^ (unsigned)(((int)u >> 31) | 0x80000000);
}
__device__ __forceinline__ float decf(unsigned e) {
    return (e & 0x80000000u) ? __uint_as_float(e ^ 0x80000000u)
                             : __uint_as_float(~e);
}
#define ENC_NEGINF 0x007FFFFFu  // encf(-inf)

// ---------- K0: q_src/q_dst = lin_w^T @ att  (fold attention through GEMM) ----------
__global__ void k_fold_att(const float* __restrict__ lin_w,
                           const float* __restrict__ att_src,
                           const float* __restrict__ att_dst,
                           float* __restrict__ q_src, float* __restrict__ q_dst) {
    int i = threadIdx.x;                 // 72 = HEADS*GAT_IN
    if (i >= HEADS * GAT_IN) return;
    int hd = i / GAT_IN, k = i % GAT_IN;
    float as = 0.f, ad = 0.f;
    for (int c = 0; c < C_DIM; ++c) {
        float w = lin_w[(size_t)(hd * C_DIM + c) * GAT_IN + k];
        as += w * att_src[hd * C_DIM + c];
        ad += w * att_dst[hd * C_DIM + c];
    }
    q_src[i] = as; q_dst[i] = ad;
}

// ---------- K1: init accumulators ----------
__global__ void k_init(float* __restrict__ gat, unsigned* __restrict__ amax,
                       float* __restrict__ denom, int N) {
    int i = blockIdx.x * blockDim.x + threadIdx.x;
    if (i < N * C_DIM) gat[i] = 0.f;
    if (i < N * HEADS) { amax[i] = ENC_NEGINF; denom[i] = 0.f; }
}

// ---------- K2: h[N,1152] = xf[N,24] @ lin_w.T  via V_WMMA_F32_16X16X4_F32 ----------
// One wave computes one 16x16 tile of h; K=24 as 6 chained K=4 WMMAs (fp32, full precision).
__global__ __launch_bounds__(256) void k_lin_wmma(const float* __restrict__ xf,
                                                  const float* __restrict__ lin_w,
                                                  float* __restrict__ h,
                                                  int Nn, int MT) {
    const int lane = threadIdx.x & 31;
    const int wid  = (int)((blockIdx.x * blockDim.x + threadIdx.x) >> 5);
    const int NT = HC / 16;                       // 72 column tiles
    if (wid >= MT * NT) return;                   // wave-uniform exit (EXEC all-1 at WMMA)
    const int tile_m = wid % MT;
    const int tile_n = wid / MT;                  // waves in a block share tile_n -> lin_w reuse

    // A frag (16x4 f32): lanes 0-15 = rows M, K={0,1}; lanes 16-31 = rows M, K={2,3}
    // B frag (4x16 f32): lanes 0-15 = cols N, K={0,1}; lanes 16-31 = cols N, K={2,3}
    const int mrow  = tile_m * 16 + (lane & 15);
    const int mr    = mrow < Nn ? mrow : (Nn - 1);          // clamp load, keep EXEC full
    const int ncol  = tile_n * 16 + (lane & 15);
    const int khalf = (lane >> 4) * 2;                      // 0 or 2
    const float* arow = xf    + (size_t)mr   * GAT_IN;
    const float* brow = lin_w + (size_t)ncol * GAT_IN;      // B[k][n] = lin_w[n][k]

    v8f acc = {};
#pragma unroll
    for (int kk = 0; kk < 6; ++kk) {
        const int k0 = kk * 4 + khalf;
        v2f a, b;
        a.x = arow[k0]; a.y = arow[k0 + 1];
        b.x = brow[k0]; b.y = brow[k0 + 1];
        acc = __builtin_amdgcn_wmma_f32_16x16x4_f32(false, a, false, b,
                                                    (short)0, acc, false, false);
    }
    // D layout: lane L holds column (L&15); VGPR v holds row v + 8*(L>>4)
    const int mbase = tile_m * 16 + 8 * (lane >> 4);
    float* hp = h + (size_t)mbase * HC + ncol;
    if (tile_m * 16 + 16 <= Nn) {
        // full tile (common path): straight-line 8 stores, no exec juggling
#pragma unroll
        for (int v = 0; v < 8; ++v) hp[(size_t)v * HC] = acc[v];
    } else {
#pragma unroll
        for (int v = 0; v < 8; ++v)
            if (mbase + v < Nn) hp[(size_t)v * HC] = acc[v];
    }
}

// ---------- K3: attention logits a_src/a_dst[N,3] = xf @ q ----------
__global__ void k_alpha_node(const float* __restrict__ xf,
                             const float* __restrict__ q_src,
                             const float* __restrict__ q_dst,
                             float* __restrict__ a_src, float* __restrict__ a_dst, int N) {
    int i = blockIdx.x * blockDim.x + threadIdx.x;
    if (i >= N * HEADS) return;
    int n = i / HEADS, hd = i % HEADS;
    float as = 0.f, ad = 0.f;
    const float* xr = xf + (size_t)n * GAT_IN;
#pragma unroll
    for (int k = 0; k < GAT_IN; ++k) {
        as += xr[k] * q_src[hd * GAT_IN + k];
        ad += xr[k] * q_dst[hd * GAT_IN + k];
    }
    a_src[i] = as; a_dst[i] = ad;
}

__device__ __forceinline__ void edge_sd(const long long* ei, int e, int E, int& s, int& d) {
    if (e < E) { s = (int)ei[e]; d = (int)ei[E + e]; }
    else       { s = d = e - E; }                    // self loops appended
}

// ---------- K4: pass 1 — segment max ----------
__global__ void k_edge_max(const long long* __restrict__ ei,
                           const float* __restrict__ a_src, const float* __restrict__ a_dst,
                           unsigned* __restrict__ amax, int E, int E2) {
    int e = blockIdx.x * blockDim.x + threadIdx.x;
    if (e >= E2) return;
    int s, d; edge_sd(ei, e, E, s, d);
#pragma unroll
    for (int hd = 0; hd < HEADS; ++hd) {
        float a = a_src[s * HEADS + hd] + a_dst[d * HEADS + hd];
        a = a > 0.f ? a : NEG_SLOPE * a;
        atomicMax(&amax[d * HEADS + hd], encf(a));
    }
}

// ---------- K5: pass 2 — exp + segment sum ----------
__global__ void k_edge_exp(const long long* __restrict__ ei,
                           const float* __restrict__ a_src, const float* __restrict__ a_dst,
                           const unsigned* __restrict__ amax,
                           float* __restrict__ ea, float* __restrict__ denom, int E, int E2) {
    int e = blockIdx.x * blockDim.x + threadIdx.x;
    if (e >= E2) return;
    int s, d; edge_sd(ei, e, E, s, d);
#pragma unroll
    for (int hd = 0; hd < HEADS; ++hd) {
        float a = a_src[s * HEADS + hd] + a_dst[d * HEADS + hd];
        a = a > 0.f ? a : NEG_SLOPE * a;
        float v = __expf(a - decf(amax[d * HEADS + hd]));
        ea[(size_t)e * HEADS + hd] = v;
        atomicAdd(&denom[d * HEADS + hd], v);
    }
}

// ---------- K6: pass 3 — weighted gather/scatter (the L2/atomic-bound phase) ----------
// One 96-thread block per edge; each thread owns 4 consecutive channels -> the
// dominant gather of h[src] is 3 x global_load_b128 per thread. Heads are summed
// in-register with the 1/3 head-mean folded in -> 4 atomicAdds per thread.
__global__ __launch_bounds__(96) void k_edge_scatter(const long long* __restrict__ ei,
                                                     const float* __restrict__ h,
                                                     const float* __restrict__ ea,
                                                     const float* __restrict__ denom,
                                                     float* __restrict__ gat, int E) {
    int e = blockIdx.x;
    int s, d; edge_sd(ei, e, E, s, d);
    const float inv3 = 1.0f / (float)HEADS;
    float w0 = ea[(size_t)e * HEADS + 0] / denom[d * HEADS + 0] * inv3;
    float w1 = ea[(size_t)e * HEADS + 1] / denom[d * HEADS + 1] * inv3;
    float w2 = ea[(size_t)e * HEADS + 2] / denom[d * HEADS + 2] * inv3;
    const float4* hs = (const float4*)(h + (size_t)s * HC);   // rows are 16B-aligned
    const int c4 = threadIdx.x;                               // 0..95 covers 384 channels
    float4 h0 = hs[c4];
    float4 h1 = hs[(C_DIM / 4) + c4];
    float4 h2 = hs[2 * (C_DIM / 4) + c4];
    float* gd = gat + (size_t)d * C_DIM + c4 * 4;
    atomicAdd(gd + 0, w0 * h0.x + w1 * h1.x + w2 * h2.x);
    atomicAdd(gd + 1, w0 * h0.y + w1 * h1.y + w2 * h2.y);
    atomicAdd(gd + 2, w0 * h0.z + w1 * h1.z + w2 * h2.z);
    atomicAdd(gd + 3, w0 * h0.w + w1 * h1.w + w2 * h2.w);
}

// ---------- K7: GRU, 12 steps in one launch; 8 nodes/block, lane = hidden unit ----------
__global__ __launch_bounds__(256) void k_gru(const float* __restrict__ gat,
                                             const float* __restrict__ gat_bias,
                                             const float* __restrict__ w_ih,
                                             const float* __restrict__ w_hh,
                                             const float* __restrict__ b_ih,
                                             const float* __restrict__ b_hh,
                                             float* __restrict__ gru_out, int N) {
    __shared__ float sWih[HID * 96];   // transposed: [k][j] -> conflict-free
    __shared__ float sWhh[HID * 96];
    __shared__ float sb[192];
    __shared__ float sx[8][HID];
    __shared__ float sh[8][HID];
    for (int i = threadIdx.x; i < 96 * HID; i += 256) {
        int j = i / HID, k = i % HID;
        sWih[k * 96 + j] = w_ih[i];
        sWhh[k * 96 + j] = w_hh[i];
    }
    if (threadIdx.x < 96)  sb[threadIdx.x] = b_ih[threadIdx.x];
    else if (threadIdx.x < 192) sb[threadIdx.x] = b_hh[threadIdx.x - 96];
    const int ln = threadIdx.x & 31;
    const int nd = threadIdx.x >> 5;
    const int node = blockIdx.x * 8 + nd;
    const bool ok = node < N;
    sh[nd][ln] = 0.f;
    __syncthreads();
    for (int t = 0; t < T_HIS; ++t) {
        if (ok) sx[nd][ln] = gat[(size_t)node * (T_HIS * HID) + t * HID + ln]
                           + gat_bias[t * HID + ln];
        __syncthreads();
        float ir = sb[ln], iz = sb[32 + ln], in_ = sb[64 + ln];
        float hr = sb[96 + ln], hz = sb[128 + ln], hn = sb[160 + ln];
#pragma unroll 8
        for (int k = 0; k < HID; ++k) {
            float xk = sx[nd][k], hk = sh[nd][k];
            const float* wi = &sWih[k * 96];
            const float* wh = &sWhh[k * 96];
            ir += wi[ln] * xk;  iz += wi[32 + ln] * xk;  in_ += wi[64 + ln] * xk;
            hr += wh[ln] * hk;  hz += wh[32 + ln] * hk;  hn  += wh[64 + ln] * hk;
        }
        float r  = 1.f / (1.f + __expf(-(ir + hr)));
        float z  = 1.f / (1.f + __expf(-(iz + hz)));
        float nn = tanhf(in_ + r * hn);
        float hp = sh[nd][ln];
        float hnew = (1.f - z) * nn + z * hp;
        __syncthreads();
        sh[nd][ln] = hnew;
        if (ok) gru_out[(size_t)node * (T_HIS * HID) + t * HID + ln] = hnew;
        __syncthreads();
    }
}

// ---------- K8: p1 (H->1) then p2 (12->12) projections ----------
__global__ void k_proj(const float* __restrict__ gru_out,
                       const float* __restrict__ p1_w, const float* __restrict__ p1_b,
                       const float* __restrict__ p2_w, const float* __restrict__ p2_b,
                       float* __restrict__ out, int N) {
    int n = blockIdx.x * blockDim.x + threadIdx.x;
    if (n >= N) return;
    float o1[T_HIS];
#pragma unroll
    for (int t = 0; t < T_HIS; ++t) {
        float a = p1_b[0];
        const float* g = gru_out + (size_t)n * (T_HIS * HID) + t * HID;
#pragma unroll
        for (int k = 0; k < HID; ++k) a += g[k] * p1_w[k];
        o1[t] = a;
    }
#pragma unroll
    for (int tp = 0; tp < T_HIS; ++tp) {
        float a = p2_b[tp];
#pragma unroll
        for (int t = 0; t < T_HIS; ++t) a += o1[t] * p2_w[tp * T_HIS + t];
        out[(size_t)n * T_HIS + tp] = a;
    }
}

static inline size_t alignup(size_t x) { return (x + 255) & ~(size_t)255; }

extern "C" void kernel_launch(void* const* d_in, const int* in_sizes, int n_in,
                              void* d_out, int out_size, void* d_ws, size_t ws_size,
                              hipStream_t stream) {
    const float*     xf       = (const float*)d_in[0];
    const long long* ei       = (const long long*)d_in[1];
    const float*     lin_w    = (const float*)d_in[2];
    const float*     att_src  = (const float*)d_in[3];
    const float*     att_dst  = (const float*)d_in[4];
    const float*     gat_bias = (const float*)d_in[5];
    const float*     w_ih     = (const float*)d_in[6];
    const float*     w_hh     = (const float*)d_in[7];
    const float*     b_ih     = (const float*)d_in[8];
    const float*     b_hh     = (const float*)d_in[9];
    const float*     p1_w     = (const float*)d_in[10];
    const float*     p1_b     = (const float*)d_in[11];
    const float*     p2_w     = (const float*)d_in[12];
    const float*     p2_b     = (const float*)d_in[13];
    float*           out      = (float*)d_out;

    const int N  = in_sizes[0] / (T_HIS * 2);   // his*in_dim = 24
    const int E  = in_sizes[1] / 2;
    const int E2 = E + N;                        // with self-loops

    // workspace layout
    char* ws = (char*)d_ws;
    size_t off = 0;
    float*    h      = (float*)(ws + off); off = alignup(off + (size_t)N * HC * 4);
    float*    q_src  = (float*)(ws + off); off = alignup(off + HEADS * GAT_IN * 4);
    float*    q_dst  = (float*)(ws + off); off = alignup(off + HEADS * GAT_IN * 4);
    float*    a_src  = (float*)(ws + off); off = alignup(off + (size_t)N * HEADS * 4);
    float*    a_dst  = (float*)(ws + off); off = alignup(off + (size_t)N * HEADS * 4);
    unsigned* amax   = (unsigned*)(ws + off); off = alignup(off + (size_t)N * HEADS * 4);
    float*    denom  = (float*)(ws + off); off = alignup(off + (size_t)N * HEADS * 4);
    float*    ea     = (float*)(ws + off); off = alignup(off + (size_t)E2 * HEADS * 4);
    float*    gat    = (float*)(ws + off); off = alignup(off + (size_t)N * C_DIM * 4);
    float*    gruo   = (float*)(ws + off); off = alignup(off + (size_t)N * C_DIM * 4);

    k_fold_att<<<1, 128, 0, stream>>>(lin_w, att_src, att_dst, q_src, q_dst);

    int init_n = N * C_DIM;
    k_init<<<(init_n + 255) / 256, 256, 0, stream>>>(gat, amax, denom, N);

    const int MT = (N + 15) / 16;
    const int waves = MT * (HC / 16);
    k_lin_wmma<<<(waves + 7) / 8, 256, 0, stream>>>(xf, lin_w, h, N, MT);

    k_alpha_node<<<(N * HEADS + 255) / 256, 256, 0, stream>>>(xf, q_src, q_dst, a_src, a_dst, N);

    k_edge_max<<<(E2 + 255) / 256, 256, 0, stream>>>(ei, a_src, a_dst, amax, E, E2);
    k_edge_exp<<<(E2 + 255) / 256, 256, 0, stream>>>(ei, a_src, a_dst, amax, ea, denom, E, E2);
    k_edge_scatter<<<E2, 96, 0, stream>>>(ei, h, ea, denom, gat, E);

    k_gru<<<(N + 7) / 8, 256, 0, stream>>>(gat, gat_bias, w_ih, w_hh, b_ih, b_hh, gruo, N);

    k_proj<<<(N + 255) / 256, 256, 0, stream>>>(gruo, p1_w, p1_b, p2_w, p2_b, out, N);

    (void)n_in; (void)out_size; (void)ws_size;
}